// N2NMP_61211873903184
// MI455X (gfx1250) — compile-verified
//
#include <hip/hip_runtime.h>

typedef float v2f __attribute__((ext_vector_type(2)));
typedef float v8f __attribute__((ext_vector_type(8)));

#define WMMA4(a, b, c) \
  __builtin_amdgcn_wmma_f32_16x16x4_f32(false, (a), false, (b), (short)0, (c), false, false)

namespace {
constexpr int XS_FLOATS = 640 * 100;                  // X half-slab in LDS (stride 100)
constexpr int QK_STRIDE = 113;                        // odd stride: conflict-free
constexpr int P_STRIDE  = 101;                        // odd stride: conflict-free
constexpr int QC_OFF    = XS_FLOATS;                  // Q chunk [64][113]
constexpr int KC_OFF    = XS_FLOATS + 64 * QK_STRIDE; // K chunk [64][113]
constexpr int PS_OFF    = XS_FLOATS;                  // P (softmax) [112][101] (reuses QK region)
constexpr int VC_OFF    = XS_FLOATS + 112 * P_STRIDE; // V chunk [64][101]
constexpr int LDS_FLOATS = XS_FLOATS + 112 * P_STRIDE + 64 * P_STRIDE; // 81776 f = 327104 B
}

struct ModP {
  const float* wq; const float* wk; const float* wv;
  const float* bq; const float* bk; const float* bv;
  int cin;    // 640 or 1280
  int choff;  // channel offset into flattened [3200] x channels
};
struct AllP { ModP m[4]; };

__device__ __forceinline__ v8f vz8() {
  v8f z;
#pragma unroll
  for (int i = 0; i < 8; ++i) z[i] = 0.0f;
  return z;
}

__global__ __launch_bounds__(256)
void fused_limb_attn(const float* __restrict__ x, AllP prm, float* __restrict__ out) {
  __shared__ float sm[LDS_FLOATS];

  const int b    = blockIdx.x;          // batch 0..255
  const int mod  = blockIdx.y;          // module 0..3
  const ModP mp  = prm.m[mod];
  const int tid  = threadIdx.x;
  const int lane = tid & 31;
  const int wave = tid >> 5;            // 0..7
  const int l16  = lane & 15;
  const int hi   = lane >> 4;           // 0 or 1 (upper half-wave)
  const int ksel = hi * 2;              // K-pair base per f32-WMMA A/B layout

  // Edge handling for the last column tile (cols 96..111): clamped index + 0/1 mask
  // instead of a predicated load -> no EXEC save/restore inside the WMMA loops.
  const int   col6 = (96 + l16 < 100) ? (96 + l16) : 99;
  const float m6   = (96 + l16 < 100) ? 1.0f : 0.0f;

  const float* xb = x + (size_t)b * 320000 + (size_t)mp.choff * 100;
  const int halves = mp.cin / 640;      // 1 (640) or 2 (1280)

  // ---------- Phase 1: S = (Wq x + bq)^T (Wk x + bk), 49 register tiles ----------
  v8f sacc[7];
#pragma unroll
  for (int i = 0; i < 7; ++i) sacc[i] = vz8();

  for (int h = 0; h < halves; ++h) {
    __syncthreads();
    {   // stage X half (contiguous 640*100 floats) -> LDS
      const float4* src = (const float4*)(xb + (size_t)h * XS_FLOATS);
      float4* dst = (float4*)sm;
      for (int i = tid; i < XS_FLOATS / 4; i += 256) dst[i] = src[i];
    }
    __syncthreads();
    const int kbase = h * 640;

    for (int c0 = 0; c0 < 640; c0 += 64) {
      // --- GEMM1: waves 0..3 -> Q strip, waves 4..7 -> K strip (16 rows each) ---
      const bool  isK  = (wave >= 4);
      const int   mt   = wave & 3;
      const float* W   = isK ? mp.wk : mp.wq;
      const float* bia = isK ? mp.bk : mp.bq;
      const int   rowA = c0 + mt * 16 + l16;
      const float* wrow = W + (size_t)rowA * mp.cin + kbase;

      if (c0 + 64 < 640)   // warm L0/L2 for next chunk's weight strip
        __builtin_prefetch(wrow + (size_t)64 * mp.cin, 0, 0);

      v8f qa[7];
#pragma unroll
      for (int nt = 0; nt < 7; ++nt) qa[nt] = vz8();

#pragma unroll 2
      for (int kt = 0; kt < 160; ++kt) {
        const int k0 = kt * 4;
        const int r0 = (k0 + ksel) * 100;
        const int r1 = r0 + 100;
        v2f a;
        a.x = wrow[k0 + ksel];
        a.y = wrow[k0 + ksel + 1];
#pragma unroll
        for (int nt = 0; nt < 6; ++nt) {   // cols 0..95: provably in-range
          v2f bb;
          bb.x = sm[r0 + nt * 16 + l16];
          bb.y = sm[r1 + nt * 16 + l16];
          qa[nt] = WMMA4(a, bb, qa[nt]);
        }
        {   // cols 96..111: clamped load * mask (branch-free)
          v2f bb;
          bb.x = sm[r0 + col6] * m6;
          bb.y = sm[r1 + col6] * m6;
          qa[6] = WMMA4(a, bb, qa[6]);
        }
      }
      __syncthreads();   // previous chunk's S-GEMM reads are done in all waves

      // store Q/K chunk (+bias) to LDS: [64 rows c][113-stride cols l]
      {
        float* dstQK = sm + (isK ? KC_OFF : QC_OFF);
#pragma unroll
        for (int r = 0; r < 8; ++r) {
          const int ml = mt * 16 + r + 8 * hi;
          const float bv_ = bia[c0 + ml];
#pragma unroll
          for (int nt = 0; nt < 7; ++nt)
            dstQK[ml * QK_STRIDE + nt * 16 + l16] = qa[nt][r] + bv_;
        }
      }
      __syncthreads();

      // --- S-GEMM: sacc += Qc^T(112x64) * Kc(64x112), tiles t%8==wave ---
      const float* Qc = sm + QC_OFF;
      const float* Kc = sm + KC_OFF;
      for (int i = 0; i < 7; ++i) {
        const int t = wave + 8 * i;
        if (t >= 49) break;
        const int lt  = t / 7;
        const int mt2 = t % 7;
        v8f acc = sacc[i];
#pragma unroll 4
        for (int kt = 0; kt < 16; ++kt) {
          const int k0 = kt * 4;
          v2f a, bb;
          a.x  = Qc[(k0 + ksel) * QK_STRIDE + lt * 16 + l16];
          a.y  = Qc[(k0 + ksel + 1) * QK_STRIDE + lt * 16 + l16];
          bb.x = Kc[(k0 + ksel) * QK_STRIDE + mt2 * 16 + l16];
          bb.y = Kc[(k0 + ksel + 1) * QK_STRIDE + mt2 * 16 + l16];
          acc = WMMA4(a, bb, acc);
        }
        sacc[i] = acc;
      }
    }
  }

  // ---------- Phase 2: softmax over m (axis -1), P stored [112][101] ----------
  __syncthreads();
  {
    float* Ps = sm + PS_OFF;
    for (int i = 0; i < 7; ++i) {
      const int t = wave + 8 * i;
      if (t >= 49) break;
      const int lt  = t / 7;
      const int mt2 = t % 7;
#pragma unroll
      for (int r = 0; r < 8; ++r) {
        const int l  = lt * 16 + r + 8 * hi;
        const int m_ = mt2 * 16 + l16;
        if (m_ < 100) Ps[l * P_STRIDE + m_] = sacc[i][r];
      }
    }
  }
  __syncthreads();
  {
    float* Ps = sm + PS_OFF;
    if (tid < 100) {
      float* row = Ps + tid * P_STRIDE;
      float mx = row[0];
      for (int m_ = 1; m_ < 100; ++m_) mx = fmaxf(mx, row[m_]);
      float s = 0.0f;
      for (int m_ = 0; m_ < 100; ++m_) { float e = __expf(row[m_] - mx); row[m_] = e; s += e; }
      const float inv = 1.0f / s;
      for (int m_ = 0; m_ < 100; ++m_) row[m_] *= inv;
    } else if (tid < 112) {
      float* row = Ps + tid * P_STRIDE;
      for (int m_ = 0; m_ < 100; ++m_) row[m_] = 0.0f;   // padded l-rows contribute 0
    }
  }
  __syncthreads();

  // ---------- Phase 3: out = (Wv x + bv) @ P^T ----------
  float* outB = out + ((size_t)mod * 256 + (size_t)b) * 64000;
  const float* Ps = sm + PS_OFF;
  const float* Vc = sm + VC_OFF;

  for (int h = 0; h < halves; ++h) {
    __syncthreads();
    if (halves > 1) {   // (halves==1: X is still resident from phase 1)
      const float4* src = (const float4*)(xb + (size_t)h * XS_FLOATS);
      float4* dst = (float4*)sm;
      for (int i = tid; i < XS_FLOATS / 4; i += 256) dst[i] = src[i];
      __syncthreads();
    }
    const int kbase = h * 640;

    for (int c0 = 0; c0 < 640; c0 += 64) {
      // --- GEMM-V: strip mtv = wave&3; waves<4 do nt 0..3, waves>=4 do nt 4..6 ---
      const int mtv  = wave & 3;
      const int ntLo = (wave < 4) ? 0 : 4;
      const int ntPl = (wave < 4) ? 4 : 6;   // plain (in-range) tiles
      const float* wrow = mp.wv + (size_t)(c0 + mtv * 16 + l16) * mp.cin + kbase;

      if (c0 + 64 < 640)
        __builtin_prefetch(wrow + (size_t)64 * mp.cin, 0, 0);

      v8f va[4];
#pragma unroll
      for (int i = 0; i < 4; ++i) va[i] = vz8();

#pragma unroll 2
      for (int kt = 0; kt < 160; ++kt) {
        const int k0 = kt * 4;
        const int r0 = (k0 + ksel) * 100;
        const int r1 = r0 + 100;
        v2f a;
        a.x = wrow[k0 + ksel];
        a.y = wrow[k0 + ksel + 1];
        for (int nt = ntLo; nt < ntPl; ++nt) {
          v2f bb;
          bb.x = sm[r0 + nt * 16 + l16];
          bb.y = sm[r1 + nt * 16 + l16];
          va[nt - ntLo] = WMMA4(a, bb, va[nt - ntLo]);
        }
        if (wave >= 4) {   // wave-uniform branch: EXEC stays all-ones per wave
          v2f bb;
          bb.x = sm[r0 + col6] * m6;
          bb.y = sm[r1 + col6] * m6;
          va[2] = WMMA4(a, bb, va[2]);
        }
      }
      __syncthreads();   // previous chunk's out-GEMM reads of Vc are done

      {   // store V chunk (+bias), only cols < 100 (K of out-GEMM stops at 100)
        float* VcW = sm + VC_OFF;
        const int ntHi = (wave < 4) ? 4 : 7;
        for (int nt = ntLo; nt < ntHi; ++nt) {
          const int colg = nt * 16 + l16;
          if (colg < 100) {
#pragma unroll
            for (int r = 0; r < 8; ++r) {
              const int ml = mtv * 16 + r + 8 * hi;
              VcW[ml * P_STRIDE + colg] = va[nt - ntLo][r] + mp.bv[c0 + ml];
            }
          }
        }
      }
      __syncthreads();

      // --- out-GEMM: out[c0+64 x 112] = Vc(64x100) * P^T(100x112); tiles t%8==wave ---
      for (int i = 0; i < 4; ++i) {
        const int t = wave + 8 * i;
        if (t >= 28) break;
        const int mtA = t / 7;
        const int ntA = t % 7;
        v8f acc;
        if (h == 0) {
          acc = vz8();
        } else {
#pragma unroll
          for (int r = 0; r < 8; ++r) {   // accumulate thigh half 1 onto half 0
            const int c   = c0 + mtA * 16 + r + 8 * hi;
            const int col = ntA * 16 + l16;
            acc[r] = (col < 100) ? outB[c * 100 + col] : 0.0f;
          }
        }
#pragma unroll 5
        for (int kt = 0; kt < 25; ++kt) {   // K = 100 exactly
          const int k0 = kt * 4;
          v2f a, bb;
          a.x  = Vc[(mtA * 16 + l16) * P_STRIDE + k0 + ksel];
          a.y  = Vc[(mtA * 16 + l16) * P_STRIDE + k0 + ksel + 1];
          bb.x = Ps[(ntA * 16 + l16) * P_STRIDE + k0 + ksel];      // B[j][i] = P[i][j]
          bb.y = Ps[(ntA * 16 + l16) * P_STRIDE + k0 + ksel + 1];
          acc = WMMA4(a, bb, acc);
        }
#pragma unroll
        for (int r = 0; r < 8; ++r) {
          const int c   = c0 + mtA * 16 + r + 8 * hi;
          const int col = ntA * 16 + l16;
          if (col < 100) outB[c * 100 + col] = acc[r];
        }
      }
    }
  }
}

extern "C" void kernel_launch(void* const* d_in, const int* in_sizes, int n_in,
                              void* d_out, int out_size, void* d_ws, size_t ws_size,
                              hipStream_t stream) {
  (void)in_sizes; (void)n_in; (void)out_size; (void)d_ws; (void)ws_size;
  const float* x = (const float*)d_in[0];

  static const int cins[4]  = {640, 1280, 640, 640};          // arm, thigh, foot, trunk
  static const int choff[4] = {8 * 320, 2 * 320, 6 * 320, 0}; // plane-start * 320

  AllP prm;
  for (int m = 0; m < 4; ++m) {
    const int base = 1 + m * 6;   // per module: wq, wk, wv, bq, bk, bv
    prm.m[m].wq = (const float*)d_in[base + 0];
    prm.m[m].wk = (const float*)d_in[base + 1];
    prm.m[m].wv = (const float*)d_in[base + 2];
    prm.m[m].bq = (const float*)d_in[base + 3];
    prm.m[m].bk = (const float*)d_in[base + 4];
    prm.m[m].bv = (const float*)d_in[base + 5];
    prm.m[m].cin   = cins[m];
    prm.m[m].choff = choff[m];
  }

  dim3 grid(256, 4);
  dim3 block(256);
  fused_limb_attn<<<grid, block, 0, stream>>>(x, prm, (float*)d_out);
}